// MultiScaleRoIAlign_48189533061803
// MI455X (gfx1250) — compile-verified
//
#include <hip/hip_runtime.h>
#include <cstdint>

// MultiScaleRoIAlign for MI455X (gfx1250).
// feat0 [2,256,200,200], feat1 [2,256,100,100], feat2 [2,256,50,50],
// feat3 [2,256,25,25], proposals [2,512,4] -> out [1024,256,7,7] f32.
//
// Gather-latency-bound (FLOPs ~0.4G, features ~L2-resident in 192MB L2).
// Design: 2 blocks per box (channel halves) for 2x resident waves,
// separable bilinear prep in LDS, bin-minor lane mapping so each wave's
// 32 gathers hit a handful of cachelines, LDS-staged output flushed with
// CDNA5 async LDS->global b128 stores (ASYNCcnt).

#define NBOX         1024
#define NCH          256
#define CH_SPLIT     2
#define CH_PER_BLK   (NCH / CH_SPLIT)            // 128
#define POUT         7
#define NBINS        49                          // 7*7
#define NSAMP        14                          // OUT*SR samples per axis
#define OUT_PER_BOX  (NCH * NBINS)               // 12544 floats
#define OUTS_PER_BLK (CH_PER_BLK * NBINS)        // 6272 floats = 25088 B

__global__ __launch_bounds__(256) void msroi_align_kernel(
    const float* __restrict__ f0, const float* __restrict__ f1,
    const float* __restrict__ f2, const float* __restrict__ f3,
    const float* __restrict__ props, float* __restrict__ out)
{
    __shared__ float sOut[OUTS_PER_BLK];         // 25088 B staging tile
    __shared__ int   sYl[NSAMP], sYh[NSAMP], sXl[NSAMP], sXh[NSAMP];
    __shared__ float sLy[NSAMP], sLx[NSAMP];
    __shared__ int   sVy[NSAMP], sVx[NSAMP];

    const int n    = blockIdx.x >> 1;            // box id, 0..1023
    const int half = blockIdx.x & 1;             // channel half
    const int tid  = threadIdx.x;                // 0..255

    // ---- box-level prep (uniform; compiler scalarizes onto s_*_f32) ----
    const float4 box  = reinterpret_cast<const float4*>(props)[n]; // x0,y0,x1,y1
    const int    bimg = n >> 9;                                    // n / 512

    const float bw = box.z - box.x;
    const float bh = box.w - box.y;
    const float s  = sqrtf(bw * bh);
    float lf = floorf(4.0f + log2f(s * (1.0f / 224.0f) + 1e-6f));
    lf = fminf(fmaxf(lf, 2.0f), 5.0f);
    const int lvl = (int)lf - 2;                                   // 0..3

    const float* feat = (lvl == 0) ? f0 : (lvl == 1) ? f1 : (lvl == 2) ? f2 : f3;
    const int   H     = 200 >> lvl;                                // 200,100,50,25 (W==H)
    const float scale = 0.25f / (float)(1 << lvl);

    // ---- separable sample prep: 14 y-records + 14 x-records into LDS ----
    if (tid < 2 * NSAMP) {
        const bool  isX = tid >= NSAMP;
        const int   k   = isX ? (tid - NSAMP) : tid;               // 0..13
        const float c0  = (isX ? box.x : box.y) * scale;
        const float c1  = (isX ? box.z : box.w) * scale;
        const float roi = fmaxf(c1 - c0, 1.0f);
        const float bsz = roi * (1.0f / POUT);
        const float off   = (float)(k >> 1) + ((float)(k & 1) + 0.5f) * 0.5f;
        const float coord = c0 + off * bsz;

        const bool  valid = (coord >= -1.0f) && (coord <= (float)H);
        const float cc    = fmaxf(coord, 0.0f);
        const int   lo0   = (int)floorf(cc);
        const bool  edge  = (lo0 >= H - 1);
        const int   lo    = edge ? (H - 1) : lo0;
        const int   hi    = edge ? (H - 1) : lo0 + 1;
        const float frac  = edge ? 0.0f : (cc - (float)lo0);

        if (!isX) { sYl[k] = lo; sYh[k] = hi; sLy[k] = frac; sVy[k] = valid ? 1 : 0; }
        else      { sXl[k] = lo; sXh[k] = hi; sLx[k] = frac; sVx[k] = valid ? 1 : 0; }
    }
    __syncthreads();

    // ---- main gather loop: 6272 outputs / 256 threads ----
    const int    HW    = H * H;
    const float* fbase = feat + ((size_t)bimg * NCH + (size_t)half * CH_PER_BLK) * HW;

    // incremental (c,bin) for o += 256 :  256 = 5*49 + 11
    int cl  = tid / NBINS;                       // local channel 0..5
    int bin = tid - cl * NBINS;

    #pragma unroll 2
    for (int o = tid; o < OUTS_PER_BLK; o += 256) {
        const int ph = bin / POUT;
        const int pw = bin - ph * POUT;

        const float* fc = fbase + (size_t)cl * HW;
        float acc = 0.0f;

        #pragma unroll
        for (int a = 0; a < 2; ++a) {
            const int   sy = 2 * ph + a;
            const int   yl = sYl[sy], yh = sYh[sy];
            const float ly = sLy[sy];
            const float hy = 1.0f - ly;
            const int   vy = sVy[sy];
            const float* rl = fc + yl * H;
            const float* rh = fc + yh * H;

            #pragma unroll
            for (int b = 0; b < 2; ++b) {
                const int   sx = 2 * pw + b;
                const int   xl = sXl[sx], xh = sXh[sx];
                const float lx = sLx[sx];
                const float hx = 1.0f - lx;
                const int   vx = sVx[sx];

                const float v = hy * hx * rl[xl] + hy * lx * rl[xh]
                              + ly * hx * rh[xl] + ly * lx * rh[xh];
                acc += (vy & vx) ? v : 0.0f;
            }
        }
        sOut[o] = acc * 0.25f;                   // mean over SR*SR = 4 samples

        // advance (c,bin) by 256 outputs
        cl += 5; bin += 11;
        const int carry = bin >= NBINS;
        bin -= carry ? NBINS : 0;
        cl  += carry ? 1 : 0;
    }
    __syncthreads();

    // ---- CDNA5 async LDS->global flush: 1568 x 128b stores (ASYNCcnt) ----
    // GVS addressing: mem = SADDR(block's output base) + VADDR(u32 byte off).
    float* outBase = out + (size_t)n * OUT_PER_BOX + (size_t)half * OUTS_PER_BLK;
    for (int i = tid; i < OUTS_PER_BLK / 4; i += 256) {
        const uint32_t goff  = (uint32_t)(i * 16);
        // low 32 bits of a generic pointer into __shared__ == LDS byte address
        const uint32_t laddr = (uint32_t)(uintptr_t)(const void*)(&sOut[i * 4]);
        asm volatile("global_store_async_from_lds_b128 %0, %1, %2"
                     :: "v"(goff), "v"(laddr), "s"(outBase)
                     : "memory");
    }
    asm volatile("s_wait_asynccnt 0" ::: "memory");
}

extern "C" void kernel_launch(void* const* d_in, const int* in_sizes, int n_in,
                              void* d_out, int out_size, void* d_ws, size_t ws_size,
                              hipStream_t stream) {
    const float* f0    = (const float*)d_in[0];
    const float* f1    = (const float*)d_in[1];
    const float* f2    = (const float*)d_in[2];
    const float* f3    = (const float*)d_in[3];
    const float* props = (const float*)d_in[4];
    float*       out   = (float*)d_out;

    (void)in_sizes; (void)n_in; (void)out_size; (void)d_ws; (void)ws_size;

    msroi_align_kernel<<<NBOX * CH_SPLIT, 256, 0, stream>>>(f0, f1, f2, f3, props, out);
}